// AttnDecoderRNN_26603027432218
// MI455X (gfx1250) — compile-verified
//
#include <hip/hip_runtime.h>

#define H 1024
#define V 32000
#define S 4096

typedef __attribute__((ext_vector_type(2))) float v2f;
typedef __attribute__((ext_vector_type(8))) float v8f;

__device__ __forceinline__ float waveRedSum(float x) {
#pragma unroll
  for (int o = 16; o > 0; o >>= 1) x += __shfl_xor(x, o, 32);
  return x;
}

// Byte offset of a generic pointer within the wave's LDS allocation.
__device__ __forceinline__ unsigned ldsByteOff(const void* p) {
  return (unsigned)(unsigned long long)(__attribute__((address_space(3))) const char*)p;
}

// ---------------------------------------------------------------------------
// Kernel A: gx = w_ih @ x  and  gh = w_hh @ h0   (biases added in kernel B)
// one wave per output row, b128 streaming loads, wave32 shuffle reduction
// ---------------------------------------------------------------------------
__global__ void k_gru_gemv(const float* __restrict__ w_ih,
                           const float* __restrict__ w_hh,
                           const float* __restrict__ emb,
                           const int*   __restrict__ widx,
                           const float* __restrict__ h0,
                           float* __restrict__ g_ih,
                           float* __restrict__ g_hh) {
  int gw   = (int)(blockIdx.x * 8 + (threadIdx.x >> 5));   // global wave id
  int lane = threadIdx.x & 31;
  const float* Wrow;
  const float* x;
  float* dst;
  int row;
  if (gw < 3 * H) {                // w_ih @ embedding[word]
    row  = gw;
    Wrow = w_ih + (size_t)row * H;
    x    = emb + (size_t)widx[0] * H;
    dst  = g_ih;
  } else {                         // w_hh @ h0
    row  = gw - 3 * H;
    Wrow = w_hh + (size_t)row * H;
    x    = h0;
    dst  = g_hh;
  }
  float acc = 0.f;
#pragma unroll
  for (int it = 0; it < 8; ++it) {
    int c = it * 128 + lane * 4;
    float4 a = *(const float4*)(Wrow + c);
    float4 b = *(const float4*)(x + c);
    acc = fmaf(a.x, b.x, acc);
    acc = fmaf(a.y, b.y, acc);
    acc = fmaf(a.z, b.z, acc);
    acc = fmaf(a.w, b.w, acc);
  }
  acc = waveRedSum(acc);
  if (lane == 0) dst[row] = acc;
}

// ---------------------------------------------------------------------------
// Kernel B: GRU gates -> h.  Writes h to v[0:H] (workspace) and d_out h-slot.
// ---------------------------------------------------------------------------
__global__ void k_gru_gates(const float* __restrict__ g_ih,
                            const float* __restrict__ g_hh,
                            const float* __restrict__ b_ih,
                            const float* __restrict__ b_hh,
                            const float* __restrict__ h0,
                            float* __restrict__ vvec,
                            float* __restrict__ out_h) {
  int i = threadIdx.x;
  float gxr = g_ih[i]          + b_ih[i];
  float ghr = g_hh[i]          + b_hh[i];
  float gxz = g_ih[H + i]      + b_ih[H + i];
  float ghz = g_hh[H + i]      + b_hh[H + i];
  float gxn = g_ih[2 * H + i]  + b_ih[2 * H + i];
  float ghn = g_hh[2 * H + i]  + b_hh[2 * H + i];
  float r = 1.f / (1.f + __expf(-(gxr + ghr)));
  float z = 1.f / (1.f + __expf(-(gxz + ghz)));
  float n = tanhf(gxn + r * ghn);
  float h = (1.f - z) * n + z * h0[i];
  vvec[i]  = h;
  out_h[i] = h;
}

// ---------------------------------------------------------------------------
// Kernel C: u = attn_w^T @ h, split over 8 j-blocks x 4 k-blocks (fixed-order)
// ---------------------------------------------------------------------------
__global__ void k_u_part(const float* __restrict__ attn_w,
                         const float* __restrict__ hvec,
                         float* __restrict__ u_part) {
  int g    = blockIdx.x;
  int kblk = g & 3;
  int jblk = g >> 2;
  int k    = kblk * 256 + threadIdx.x;
  int j0   = jblk * 128;
  float acc = 0.f;
  for (int j = 0; j < 128; ++j)
    acc = fmaf(hvec[j0 + j], attn_w[(size_t)(j0 + j) * H + k], acc);
  u_part[jblk * H + k] = acc;
}

__global__ void k_u_reduce(const float* __restrict__ u_part,
                           float* __restrict__ u) {
  int k = threadIdx.x;
  float s = 0.f;
#pragma unroll
  for (int b = 0; b < 8; ++b) s += u_part[b * H + k];
  u[k] = s;
}

// ---------------------------------------------------------------------------
// Kernel D: scores = enc @ u  via V_WMMA_F32_16X16X4_F32.
// Wave owns 16 rows.  A: lanes 0-15 -> K={0,1}, lanes 16-31 -> K={2,3}
// (ISA 7.12.2).  B column 0 carries u[k..k+3]; other columns zero.
// D column 0: lane 0 holds M=0..7, lane 16 holds M=8..15 (VGPRs 0..7).
// ---------------------------------------------------------------------------
__global__ void k_scores_wmma(const float* __restrict__ enc,
                              const float* __restrict__ u,
                              float* __restrict__ scores) {
  int lane = threadIdx.x & 31;
  int wv   = threadIdx.x >> 5;
  int r0   = blockIdx.x * 64 + wv * 16;
  int m    = lane & 15;
  int hi   = lane >> 4;
  const float* rowp = enc + (size_t)(r0 + m) * H + 2 * hi;
  v8f acc = {0.f, 0.f, 0.f, 0.f, 0.f, 0.f, 0.f, 0.f};
  for (int k = 0; k < H; k += 4) {
    v2f a = *(const v2f*)(rowp + k);            // A fragment (b64 load)
    float b0 = u[k + 2 * hi];
    float b1 = u[k + 2 * hi + 1];
    v2f b;
    b.x = (m == 0) ? b0 : 0.f;                  // B rows 0/2 in VGPR0
    b.y = (m == 0) ? b1 : 0.f;                  // B rows 1/3 in VGPR1
    acc = __builtin_amdgcn_wmma_f32_16x16x4_f32(false, a, false, b,
                                                (short)0, acc, false, false);
  }
  if (m == 0) {
    float* dst = scores + r0 + hi * 8;
#pragma unroll
    for (int i = 0; i < 8; ++i) dst[i] = acc[i];
  }
}

// ---------------------------------------------------------------------------
// Kernel E: softmax over 4096 scores, single workgroup (1024 thr, 32 waves)
// ---------------------------------------------------------------------------
__global__ void k_softmax(const float* __restrict__ scores,
                          float* __restrict__ attn) {
  __shared__ float red[32];
  int tid = threadIdx.x, lane = tid & 31, wv = tid >> 5;
  float x0 = scores[tid];
  float x1 = scores[tid + 1024];
  float x2 = scores[tid + 2048];
  float x3 = scores[tid + 3072];
  float m = fmaxf(fmaxf(x0, x1), fmaxf(x2, x3));
#pragma unroll
  for (int o = 16; o > 0; o >>= 1) m = fmaxf(m, __shfl_xor(m, o, 32));
  if (lane == 0) red[wv] = m;
  __syncthreads();
  if (tid < 32) {
    float t = red[tid];
#pragma unroll
    for (int o = 16; o > 0; o >>= 1) t = fmaxf(t, __shfl_xor(t, o, 32));
    if (tid == 0) red[0] = t;
  }
  __syncthreads();
  m = red[0];
  __syncthreads();
  float e0 = __expf(x0 - m), e1 = __expf(x1 - m);
  float e2 = __expf(x2 - m), e3 = __expf(x3 - m);
  float s = e0 + e1 + e2 + e3;
  s = waveRedSum(s);
  if (lane == 0) red[wv] = s;
  __syncthreads();
  if (tid < 32) {
    float t = red[tid];
#pragma unroll
    for (int o = 16; o > 0; o >>= 1) t += __shfl_xor(t, o, 32);
    if (tid == 0) red[0] = t;
  }
  __syncthreads();
  float inv = 1.f / red[0];
  attn[tid]        = e0 * inv;
  attn[tid + 1024] = e1 * inv;
  attn[tid + 2048] = e2 * inv;
  attn[tid + 3072] = e3 * inv;
}

// ---------------------------------------------------------------------------
// Kernel F: context[k] = sum_i attn[i]*enc[i][k]  (enc re-read is L2-resident)
// ---------------------------------------------------------------------------
__global__ void k_ctx_part(const float* __restrict__ enc,
                           const float* __restrict__ attn,
                           float* __restrict__ c_part) {
  int iblk = blockIdx.x, k = threadIdx.x, i0 = iblk * 256;
  float acc = 0.f;
  for (int i = 0; i < 256; ++i)
    acc = fmaf(attn[i0 + i], enc[(size_t)(i0 + i) * H + k], acc);
  c_part[iblk * H + k] = acc;
}

__global__ void k_ctx_reduce(const float* __restrict__ c_part,
                             float* __restrict__ ctx) {
  int k = threadIdx.x;
  float s = 0.f;
#pragma unroll
  for (int b = 0; b < 16; ++b) s += c_part[b * H + k];
  ctx[k] = s;
}

// ---------------------------------------------------------------------------
// Kernel G: output = out_w @ [h;context] + out_b   (262 MB stream — dominant)
// v staged into LDS via CDNA5 async global->LDS copy (ASYNCcnt path),
// then one wave per row, b128 loads, wave reduction.
// ---------------------------------------------------------------------------
__global__ void k_out_gemv(const float* __restrict__ out_w,
                           const float* __restrict__ out_b,
                           const float* __restrict__ vvec,
                           float* __restrict__ out) {
  __shared__ float sv[2 * H];
  int tid = threadIdx.x;

  // async stage: 256 threads x 2 x b128 = 8 KB
  unsigned off0 = ldsByteOff(&sv[tid * 4]);
  const float* g0 = vvec + tid * 4;
  asm volatile("global_load_async_to_lds_b128 %0, %1, off"
               :: "v"(off0), "v"(g0) : "memory");
  unsigned off1 = off0 + 4096;
  const float* g1 = g0 + 1024;
  asm volatile("global_load_async_to_lds_b128 %0, %1, off"
               :: "v"(off1), "v"(g1) : "memory");
  asm volatile("s_wait_asynccnt 0" ::: "memory");
  __syncthreads();

  int wv = tid >> 5, lane = tid & 31;
  int row = blockIdx.x * 8 + wv;
  const float* W = out_w + (size_t)row * (2 * H);
  float acc = 0.f;
#pragma unroll
  for (int it = 0; it < 16; ++it) {
    int c = it * 128 + lane * 4;
    float4 a = *(const float4*)(W + c);
    float4 b = *(const float4*)(&sv[c]);
    acc = fmaf(a.x, b.x, acc);
    acc = fmaf(a.y, b.y, acc);
    acc = fmaf(a.z, b.z, acc);
    acc = fmaf(a.w, b.w, acc);
  }
  acc = waveRedSum(acc);
  if (lane == 0) out[row] = acc + out_b[row];
}

// ---------------------------------------------------------------------------
extern "C" void kernel_launch(void* const* d_in, const int* in_sizes, int n_in,
                              void* d_out, int out_size, void* d_ws, size_t ws_size,
                              hipStream_t stream) {
  const int*   widx   = (const int*)  d_in[0];
  const float* h0     = (const float*)d_in[1];   // last_hidden (1,1,H)
  const float* enc    = (const float*)d_in[2];   // (S,1,H)
  const float* emb    = (const float*)d_in[3];   // (V,H)
  const float* w_ih   = (const float*)d_in[4];
  const float* w_hh   = (const float*)d_in[5];
  const float* b_ih   = (const float*)d_in[6];
  const float* b_hh   = (const float*)d_in[7];
  const float* attn_w = (const float*)d_in[8];
  // d_in[9] (attn_b): adds a uniform constant to all scores -> cancels in softmax
  const float* out_w  = (const float*)d_in[10];
  const float* out_b  = (const float*)d_in[11];

  float* out   = (float*)d_out;          // [0, 32000)       output
  float* out_h = out + V;                // [32000, 33024)   h
  float* attn  = out + V + H;            // [33024, 37120)   attn

  float* ws     = (float*)d_ws;
  float* g_ih   = ws;                    // 3072
  float* g_hh   = ws + 3072;             // 3072
  float* vvec   = ws + 6144;             // 2048  ([h | context])
  float* u_part = ws + 8192;             // 8*1024
  float* u      = ws + 16384;            // 1024
  float* scores = ws + 17408;            // 4096
  float* c_part = ws + 21504;            // 16*1024   (end = 37888 floats)

  k_gru_gemv  <<<768, 256, 0, stream>>>(w_ih, w_hh, emb, widx, h0, g_ih, g_hh);
  k_gru_gates <<<1, 1024, 0, stream>>>(g_ih, g_hh, b_ih, b_hh, h0, vvec, out_h);
  k_u_part    <<<32, 256, 0, stream>>>(attn_w, vvec, u_part);
  k_u_reduce  <<<1, 1024, 0, stream>>>(u_part, u);
  k_scores_wmma<<<64, 128, 0, stream>>>(enc, u, scores);
  k_softmax   <<<1, 1024, 0, stream>>>(scores, attn);
  k_ctx_part  <<<16, 1024, 0, stream>>>(enc, attn, c_part);
  k_ctx_reduce<<<1, 1024, 0, stream>>>(c_part, vvec + H);
  k_out_gemv  <<<V / 8, 256, 0, stream>>>(out_w, out_b, vvec, out);
}